// AbstractNurbs_89781996355764
// MI455X (gfx1250) — compile-verified
//
#include <hip/hip_runtime.h>

// ---------------------------------------------------------------------------
// Bicubic clamped B-spline surface eval, p=3, 64x64 control net, 2001x2001 grid.
// Stage 2 (dominant FLOPs) runs on V_WMMA_F32_16X16X4_F32 (full f32 precision).
// Output (48 MB) is written exactly once; intermediates (<2.2 MB) stay in L2.
// ---------------------------------------------------------------------------

typedef __attribute__((ext_vector_type(2))) float v2f;
typedef __attribute__((ext_vector_type(8))) float v8f;

#define NEV  2001      // parameters per axis
#define NC   64        // control points per axis
#define KD   64        // GEMM K (= NC)
#define MPAD 2016      // 126 * 16   (u padded to m-tile multiple)
#define VPAD 2048      // 4 * 8 * 64 (v padded to wave n-block multiple)
#define MTILES 126     // MPAD / 16
#define NBLKY  4       // VPAD / (8 waves * 64 cols)

// clamped uniform knot vector: 4 zeros, (j-3)/61 for j=4..63, 4 ones
__device__ __forceinline__ float knotval(int j) {
  float x = (float)(j - 3) * (1.0f / 61.0f);
  return fminf(fmaxf(x, 0.0f), 1.0f);
}

// Cox–de Boor, degree 3. Returns span s (in [3,63]) and the 4 nonzero weights
// for control indices s-3..s.
__device__ __forceinline__ void cubic_basis(float t, int* span, float N[4]) {
  int s = (int)(t * 61.0f);
  s = s > 60 ? 60 : (s < 0 ? 0 : s);
  s += 3;
  float left[4], right[4];
  N[0] = 1.0f;
#pragma unroll
  for (int j = 1; j <= 3; ++j) {
    left[j]  = t - knotval(s + 1 - j);
    right[j] = knotval(s + j) - t;
    float saved = 0.0f;
#pragma unroll
    for (int r = 0; r < j; ++r) {
      float temp = N[r] / (right[r + 1] + left[j - r]);
      N[r] = saved + right[r + 1] * temp;
      saved = left[j - r] * temp;
    }
    N[j] = saved;
  }
  *span = s;
}

// Kernel 1: spans + compact Bu rows for u; dense K-major BvT[64][VPAD] for v.
__global__ void __launch_bounds__(256)
nurbs_basis_kernel(const float* __restrict__ pu, const float* __restrict__ pv,
                   int* __restrict__ su, float* __restrict__ Bu,
                   float* __restrict__ BvT) {
  int i = blockIdx.x * blockDim.x + threadIdx.x;
  if (i >= NEV) return;

  int s;
  float N[4];
  cubic_basis(pu[i], &s, N);
  su[i] = s;
  Bu[i * 4 + 0] = N[0];
  Bu[i * 4 + 1] = N[1];
  Bu[i * 4 + 2] = N[2];
  Bu[i * 4 + 3] = N[3];

  cubic_basis(pv[i], &s, N);
  int j0 = s - 3;
#pragma unroll
  for (int j = 0; j < NC; ++j) {
    float val = (j == j0)     ? N[0]
              : (j == j0 + 1) ? N[1]
              : (j == j0 + 2) ? N[2]
              : (j == j0 + 3) ? N[3]
                              : 0.0f;
    BvT[j * VPAD + i] = val;  // column-major scatter: coalesced across threads
  }
}

// Kernel 2: curve[d][u][j] = sum_a Bu[u][a] * ctrl[su[u]-3+a][j][d]
// (sparse 4-tap contraction; dense GEMM would waste 16x here)
__global__ void __launch_bounds__(256)
nurbs_curve_kernel(const float* __restrict__ ctrl, const int* __restrict__ su,
                   const float* __restrict__ Bu, float* __restrict__ curve) {
  int idx = blockIdx.x * blockDim.x + threadIdx.x;
  int u = idx >> 6;
  int j = idx & 63;
  if (u >= NEV) return;
  int s = su[u];
  float a0 = 0.0f, a1 = 0.0f, a2 = 0.0f;
#pragma unroll
  for (int a = 0; a < 4; ++a) {
    float w = Bu[u * 4 + a];
    const float* p = ctrl + ((s - 3 + a) * NC + j) * 3;
    a0 = fmaf(w, p[0], a0);
    a1 = fmaf(w, p[1], a1);
    a2 = fmaf(w, p[2], a2);
  }
  curve[0 * MPAD * KD + u * KD + j] = a0;
  curve[1 * MPAD * KD + u * KD + j] = a1;
  curve[2 * MPAD * KD + u * KD + j] = a2;
}

// Kernel 3: out[u][v][d] = sum_j curve[d][u][j] * BvT[j][v]  via f32 WMMA.
// Per wave: 16 u-rows x 64 v-cols x 3 d-planes -> 12 v8f accumulators.
// A fragment shared across the 4 n-subtiles; B fragments shared across d.
__global__ void __launch_bounds__(256)
nurbs_surf_wmma_kernel(const float* __restrict__ curve,
                       const float* __restrict__ BvT,
                       float* __restrict__ out) {
  const int lane = threadIdx.x & 31;
  const int wave = threadIdx.x >> 5;
  const int half = lane >> 4;   // lane 0-15 vs 16-31
  const int lq   = lane & 15;

  const int u0 = blockIdx.x * 16;
  const int n0 = (blockIdx.y * 8 + wave) * 64;

  v8f c[3][4] = {};

  // A 16x4 f32 layout: lane (m=lq, half) holds K = 2*half, 2*half+1
  const float* arow = curve + (u0 + lq) * KD + 2 * half;
  // B 4x16 f32 layout (mirror): lane (n=lq, half) holds K = 2*half, 2*half+1
  const float* bcol = BvT + (2 * half) * VPAD + n0 + lq;

  for (int k = 0; k < KD; k += 4) {
    v2f a[3];
#pragma unroll
    for (int d = 0; d < 3; ++d)
      a[d] = *(const v2f*)(arow + d * (MPAD * KD) + k);

    v2f b[4];
#pragma unroll
    for (int q = 0; q < 4; ++q) {
      const float* bp = bcol + k * VPAD + q * 16;
      b[q].x = bp[0];
      b[q].y = bp[VPAD];
    }

#pragma unroll
    for (int d = 0; d < 3; ++d)
#pragma unroll
      for (int q = 0; q < 4; ++q)
        c[d][q] = __builtin_amdgcn_wmma_f32_16x16x4_f32(
            /*neg_a=*/false, a[d], /*neg_b=*/false, b[q],
            /*c_mod=*/(short)0, c[d][q], /*reuse_a=*/false, /*reuse_b=*/false);
  }

  // C/D 16x16 f32: VGPR i <-> row (i + 8*half), col = lq (+ tile offsets).
  // Write the 3 d-planes contiguously so stores coalesce into dense spans.
#pragma unroll
  for (int q = 0; q < 4; ++q) {
    int v = n0 + q * 16 + lq;
    if (v >= NEV) continue;
#pragma unroll
    for (int i = 0; i < 8; ++i) {
      int u = u0 + i + half * 8;
      if (u < NEV) {
        float* o = out + ((size_t)u * NEV + v) * 3;
        o[0] = c[0][q][i];
        o[1] = c[1][q][i];
        o[2] = c[2][q][i];
      }
    }
  }
}

extern "C" void kernel_launch(void* const* d_in, const int* in_sizes, int n_in,
                              void* d_out, int out_size, void* d_ws, size_t ws_size,
                              hipStream_t stream) {
  const float* ctrl = (const float*)d_in[0];  // [64][64][3] f32
  const float* pu   = (const float*)d_in[1];  // [2001] f32
  const float* pv   = (const float*)d_in[2];  // [2001] f32
  // d_in[3] = degree (always 3; baked into the kernels)

  char* ws = (char*)d_ws;
  int*   su    = (int*)(ws);                                   //  8 KB
  float* Bu    = (float*)(ws + 8192);                          // 32 KB
  float* BvT   = (float*)(ws + 8192 + 32768);                  // 64*2048*4 = 512 KB
  float* curve = (float*)(ws + 8192 + 32768 + (size_t)KD * VPAD * 4);  // 3*2016*64*4
  float* out   = (float*)d_out;

  nurbs_basis_kernel<<<(NEV + 255) / 256, 256, 0, stream>>>(pu, pv, su, Bu, BvT);
  nurbs_curve_kernel<<<(NEV * NC + 255) / 256, 256, 0, stream>>>(ctrl, su, Bu, curve);
  nurbs_surf_wmma_kernel<<<dim3(MTILES, NBLKY), 256, 0, stream>>>(curve, BvT, out);
}